// TripGAT_46213848105116
// MI455X (gfx1250) — compile-verified
//
#include <hip/hip_runtime.h>
#include <hip/hip_bf16.h>
#include <math.h>

typedef __attribute__((ext_vector_type(2))) float v2f;
typedef __attribute__((ext_vector_type(8))) float v8f;

// ---------- ordered-float encoding for atomic max on floats ----------
__device__ __forceinline__ unsigned f2ord(float f) {
  unsigned u = __float_as_uint(f);
  return (u & 0x80000000u) ? ~u : (u | 0x80000000u);
}
__device__ __forceinline__ float ord2f(unsigned o) {
  unsigned u = (o & 0x80000000u) ? (o & 0x7FFFFFFFu) : ~o;
  return __uint_as_float(u);
}

// =====================================================================
// FP32 WMMA GEMM:  C[M,N] = A[M,K] @ B[K,N]   (row-major)
// Block: 256 threads = 8 waves. Block tile 64(M) x 128(N), K chunk 16.
// Wave w owns N sub-tile w*16 and FOUR 16x16 M sub-tiles (4 accums):
// the B fragment is loaded once per k-step and reused by 4 WMMAs.
// Requires N%128==0, K%16==0; M edge handled by clamp/predication.
// V_WMMA_F32_16X16X4_F32: A/B frag float2 (K 0/1 lanes 0-15, K 2/3
// lanes 16-31), C/D float8 (row r+8*half, col lane%16).
// =====================================================================
__global__ __launch_bounds__(256) void gat_gemm_f32(
    const float* __restrict__ A, const float* __restrict__ B,
    float* __restrict__ C, int M, int N, int K) {
  __shared__ float As[64 * 17];   // 64 rows x 16 k (pad 17: coprime w/ 64)
  __shared__ float Bs[16 * 144];  // 16 k x 128 n (pad 144: 144%64==16)

  const int t = threadIdx.x;
  const int lane = t & 31;
  const int w = t >> 5;      // 0..7 -> N sub-tile
  const int row = lane & 15;
  const int half = lane >> 4;
  const int nw = w << 4;
  const int m0 = blockIdx.y * 64;
  const int n0 = blockIdx.x * 128;

  v8f acc[4] = {{}, {}, {}, {}};
  for (int k0 = 0; k0 < K; k0 += 16) {
#pragma unroll
    for (int i = 0; i < 4; i++) {  // stage A tile 64x16
      int idx = t + i * 256;
      int r = idx >> 4, c = idx & 15;
      int gr = m0 + r;
      if (gr >= M) gr = M - 1;  // clamp (harmless dup reads at M edge)
      As[r * 17 + c] = A[(size_t)gr * K + k0 + c];
    }
#pragma unroll
    for (int i = 0; i < 8; i++) {  // stage B tile 16x128
      int idx = t + i * 256;
      int r = idx >> 7, c = idx & 127;
      Bs[r * 144 + c] = B[(size_t)(k0 + r) * N + n0 + c];
    }
    __syncthreads();
#pragma unroll
    for (int kk = 0; kk < 4; kk++) {
      const int k = kk * 4 + half * 2;
      v2f b;
      b.x = Bs[k * 144 + nw + row];
      b.y = Bs[(k + 1) * 144 + nw + row];
#pragma unroll
      for (int mt = 0; mt < 4; mt++) {
        v2f a;
        a.x = As[(mt * 16 + row) * 17 + k];
        a.y = As[(mt * 16 + row) * 17 + k + 1];
        acc[mt] = __builtin_amdgcn_wmma_f32_16x16x4_f32(
            false, a, false, b, (short)0, acc[mt], false, false);
      }
    }
    __syncthreads();
  }
  const int col = n0 + nw + row;
#pragma unroll
  for (int mt = 0; mt < 4; mt++) {
    const int rbase = m0 + mt * 16 + half * 8;
#pragma unroll
    for (int r = 0; r < 8; r++) {
      const int rr = rbase + r;
      if (rr < M) C[(size_t)rr * N + col] = acc[mt][r];
    }
  }
}

// ---------------------------------------------------------------------
// alpha_src[n,h] = sum_c H[n,h*C+c]*attS[h,c]; same for alpha_dst.
// One wave per (node, head).
// ---------------------------------------------------------------------
__global__ void gat_alpha(const float* __restrict__ Hm,
                          const float* __restrict__ attS,
                          const float* __restrict__ attD,
                          float* __restrict__ aS, float* __restrict__ aD,
                          int n_nodes, int Hh, int C) {
  const int wid = (blockIdx.x * blockDim.x + threadIdx.x) >> 5;
  const int lane = threadIdx.x & 31;
  if (wid >= n_nodes * Hh) return;
  const int hh = wid % Hh;
  const float* hp = Hm + (size_t)wid * C;  // wid = node*Hh+hh, contiguous
  float s = 0.f, d = 0.f;
  for (int c = lane; c < C; c += 32) {
    const float v = hp[c];
    s += v * attS[hh * C + c];
    d += v * attD[hh * C + c];
  }
#pragma unroll
  for (int m = 16; m >= 1; m >>= 1) {
    s += __shfl_xor(s, m, 32);
    d += __shfl_xor(d, m, 32);
  }
  if (lane == 0) { aS[wid] = s; aD[wid] = d; }
}

__device__ __forceinline__ void edge_ends(const long long* __restrict__ ei,
                                          int e, int n_edges, int& s, int& d) {
  if (e < n_edges) {
    s = (int)ei[e];
    d = (int)ei[n_edges + e];
  } else {
    s = d = e - n_edges;  // self loop
  }
}

// e = leakyrelu(aS[src]+aD[dst]); store raw e; atomic segment-max to dst.
template <int HH>
__global__ void gat_edge_lrelu_max(const long long* __restrict__ ei,
                                   int n_edges, int n_nodes,
                                   const float* __restrict__ aS,
                                   const float* __restrict__ aD,
                                   float* __restrict__ ebuf,
                                   unsigned* __restrict__ mord) {
  const int idx = blockIdx.x * blockDim.x + threadIdx.x;
  const int tot = (n_edges + n_nodes) * HH;
  if (idx >= tot) return;
  const int e = idx / HH, hh = idx % HH;  // HH is 1 or 8 -> shifts
  int s, d;
  edge_ends(ei, e, n_edges, s, d);
  float v = aS[s * HH + hh] + aD[d * HH + hh];
  v = v > 0.f ? v : 0.2f * v;
  ebuf[idx] = v;
  atomicMax(&mord[d * HH + hh], f2ord(v));
}

// ex = exp(e - m[dst]); overwrite ebuf; atomic segment-sum into denom.
template <int HH>
__global__ void gat_edge_expsum(const long long* __restrict__ ei, int n_edges,
                                int n_nodes, float* __restrict__ ebuf,
                                const unsigned* __restrict__ mord,
                                float* __restrict__ denom) {
  const int idx = blockIdx.x * blockDim.x + threadIdx.x;
  const int tot = (n_edges + n_nodes) * HH;
  if (idx >= tot) return;
  const int e = idx / HH, hh = idx % HH;
  int s, d;
  edge_ends(ei, e, n_edges, s, d);
  const float m = ord2f(mord[d * HH + hh]);
  const float ex = expf(ebuf[idx] - m);
  ebuf[idx] = ex;
  atomicAdd(&denom[d * HH + hh], ex);
}

// agg[dst,h,c] += (ex/denom[dst,h]) * Hfeat[src,h,c]. One wave per edge.
template <int HH, int CC>
__global__ void gat_aggregate(const long long* __restrict__ ei, int n_edges,
                              int n_nodes, const float* __restrict__ Hfeat,
                              const float* __restrict__ ebuf,
                              const float* __restrict__ denom,
                              float* __restrict__ agg) {
  const int e = (blockIdx.x * blockDim.x + threadIdx.x) >> 5;
  const int lane = threadIdx.x & 31;
  if (e >= n_edges + n_nodes) return;
  int s, d;
  edge_ends(ei, e, n_edges, s, d);
  constexpr int CT = HH * CC;
  const float* hs = Hfeat + (size_t)s * CT;
  float* ag = agg + (size_t)d * CT;
  // warm the gathered row (global_prefetch_b8); each lane covers CT/32 floats
  __builtin_prefetch(&hs[lane * (CT >> 5)], 0, 0);
  float aH = 0.f;
  if (lane < HH) aH = ebuf[e * HH + lane] / (denom[d * HH + lane] + 1e-16f);
#pragma unroll 4
  for (int c = lane; c < CT; c += 32) {
    const float alpha = __shfl(aH, c / CC, 32);  // c/CC -> shift
    atomicAdd(&ag[c], alpha * hs[c]);
  }
}

// x = elu(x + bias)   (in place); CT is a power of two.
__global__ void gat_elu_bias(float* __restrict__ x,
                             const float* __restrict__ bias, int n,
                             int ctmask) {
  const int i = blockIdx.x * blockDim.x + threadIdx.x;
  if (i >= n) return;
  const float v = x[i] + bias[i & ctmask];
  x[i] = v > 0.f ? v : expm1f(v);
}

// out = log_softmax(agg + bias) over C=128; one wave per node.
__global__ void gat_logsoftmax(const float* __restrict__ agg,
                               const float* __restrict__ bias,
                               float* __restrict__ out, int n_nodes, int C) {
  const int wid = (blockIdx.x * blockDim.x + threadIdx.x) >> 5;
  const int lane = threadIdx.x & 31;
  if (wid >= n_nodes) return;
  const float* row = agg + (size_t)wid * C;
  float v[4];
  float mx;
#pragma unroll
  for (int i = 0; i < 4; i++) {
    const int c = lane + i * 32;
    v[i] = row[c] + bias[c];
    mx = (i == 0) ? v[0] : fmaxf(mx, v[i]);
  }
#pragma unroll
  for (int m = 16; m >= 1; m >>= 1) mx = fmaxf(mx, __shfl_xor(mx, m, 32));
  float se = 0.f;
#pragma unroll
  for (int i = 0; i < 4; i++) se += expf(v[i] - mx);
#pragma unroll
  for (int m = 16; m >= 1; m >>= 1) se += __shfl_xor(se, m, 32);
  const float lse = logf(se);
#pragma unroll
  for (int i = 0; i < 4; i++)
    out[(size_t)wid * C + lane + i * 32] = v[i] - mx - lse;
}

// =====================================================================
extern "C" void kernel_launch(void* const* d_in, const int* in_sizes, int n_in,
                              void* d_out, int out_size, void* d_ws,
                              size_t ws_size, hipStream_t stream) {
  (void)n_in; (void)out_size; (void)ws_size;
  const float* x       = (const float*)d_in[0];
  const long long* ei  = (const long long*)d_in[1];  // int64 [2,E]
  const float* W1      = (const float*)d_in[2];
  const float* attS1   = (const float*)d_in[3];
  const float* attD1   = (const float*)d_in[4];
  const float* b1      = (const float*)d_in[5];
  const float* W2      = (const float*)d_in[6];
  const float* attS2   = (const float*)d_in[7];
  const float* attD2   = (const float*)d_in[8];
  const float* b2      = (const float*)d_in[9];

  const int F_IN = 256, H1 = 8, C1 = 128, CT1 = H1 * C1 /*1024*/, C2 = 128;
  const int N = in_sizes[0] / F_IN;  // 20000
  const int E = in_sizes[1] / 2;     // 320000
  const int ET = E + N;              // +self loops

  // ---- workspace carve (256B aligned) ----
  char* w = (char*)d_ws;
  auto carve = [&](size_t bytes) -> void* {
    void* p = (void*)w;
    w += (bytes + 255) & ~(size_t)255;
    return p;
  };
  float*    h1   = (float*)carve((size_t)N * CT1 * 4);
  float*    agg1 = (float*)carve((size_t)N * CT1 * 4);
  float*    as1  = (float*)carve((size_t)N * H1 * 4);
  float*    ad1  = (float*)carve((size_t)N * H1 * 4);
  unsigned* m1   = (unsigned*)carve((size_t)N * H1 * 4);
  float*    dn1  = (float*)carve((size_t)N * H1 * 4);
  float*    ex1  = (float*)carve((size_t)ET * H1 * 4);
  float*    h2   = (float*)carve((size_t)N * C2 * 4);
  float*    as2  = (float*)carve((size_t)N * 4);
  float*    ad2  = (float*)carve((size_t)N * 4);
  unsigned* m2   = (unsigned*)carve((size_t)N * 4);
  float*    dn2  = (float*)carve((size_t)N * 4);
  float*    ex2  = (float*)carve((size_t)ET * 4);
  float*    agg2 = (float*)carve((size_t)N * C2 * 4);

  hipMemsetAsync(m1,   0, (size_t)N * H1 * 4,  stream);
  hipMemsetAsync(dn1,  0, (size_t)N * H1 * 4,  stream);
  hipMemsetAsync(agg1, 0, (size_t)N * CT1 * 4, stream);
  hipMemsetAsync(m2,   0, (size_t)N * 4,       stream);
  hipMemsetAsync(dn2,  0, (size_t)N * 4,       stream);
  hipMemsetAsync(agg2, 0, (size_t)N * C2 * 4,  stream);

  // ===== Layer 1 =====
  dim3 g1(CT1 / 128, (N + 63) / 64);
  gat_gemm_f32<<<g1, 256, 0, stream>>>(x, W1, h1, N, CT1, F_IN);

  gat_alpha<<<(N * H1 * 32 + 255) / 256, 256, 0, stream>>>(h1, attS1, attD1,
                                                           as1, ad1, N, H1, C1);
  const int tot1 = ET * H1;
  gat_edge_lrelu_max<8><<<(tot1 + 255) / 256, 256, 0, stream>>>(ei, E, N, as1,
                                                                ad1, ex1, m1);
  gat_edge_expsum<8><<<(tot1 + 255) / 256, 256, 0, stream>>>(ei, E, N, ex1, m1,
                                                             dn1);
  gat_aggregate<8, 128><<<(ET * 32 + 255) / 256, 256, 0, stream>>>(
      ei, E, N, h1, ex1, dn1, agg1);
  gat_elu_bias<<<(N * CT1 + 255) / 256, 256, 0, stream>>>(agg1, b1, N * CT1,
                                                          CT1 - 1);

  // ===== Layer 2 =====
  dim3 g2(C2 / 128, (N + 63) / 64);
  gat_gemm_f32<<<g2, 256, 0, stream>>>(agg1, W2, h2, N, C2, CT1);

  gat_alpha<<<(N * 32 + 255) / 256, 256, 0, stream>>>(h2, attS2, attD2, as2,
                                                      ad2, N, 1, C2);
  gat_edge_lrelu_max<1><<<(ET + 255) / 256, 256, 0, stream>>>(ei, E, N, as2,
                                                              ad2, ex2, m2);
  gat_edge_expsum<1><<<(ET + 255) / 256, 256, 0, stream>>>(ei, E, N, ex2, m2,
                                                           dn2);
  gat_aggregate<1, 128><<<(ET * 32 + 255) / 256, 256, 0, stream>>>(
      ei, E, N, h2, ex2, dn2, agg2);
  gat_logsoftmax<<<(N * 32 + 255) / 256, 256, 0, stream>>>(agg2, b2,
                                                           (float*)d_out, N, C2);
}